// CABlock_76605036691835
// MI455X (gfx1250) — compile-verified
//
#include <hip/hip_runtime.h>
#include <hip/hip_bf16.h>
#include <math.h>

typedef __attribute__((ext_vector_type(2))) float v2f;
typedef __attribute__((ext_vector_type(8))) float v8f;

static constexpr int NN = 2304;   // h*w = 48*48
static constexpr int CC = 64;     // channels
static constexpr int CRr = 8;     // reduced channels
static constexpr int BB = 8;      // batch

// workspace layout (float offsets)
static constexpr long long OFF_QT1 = 0;
static constexpr long long OFF_KT1 = OFF_QT1 + (long long)BB*NN*CRr;
static constexpr long long OFF_QT2 = OFF_KT1 + (long long)BB*NN*CRr;
static constexpr long long OFF_KT2 = OFF_QT2 + (long long)BB*NN*CRr;
static constexpr long long OFF_VT1 = OFF_KT2 + (long long)BB*NN*CRr;
static constexpr long long OFF_VT2 = OFF_VT1 + (long long)BB*NN*CC;
static constexpr long long OFF_M1  = OFF_VT2 + (long long)BB*NN*CC;
static constexpr long long OFF_Z1  = OFF_M1 + (long long)BB*NN;
static constexpr long long OFF_M2  = OFF_Z1 + (long long)BB*NN;
static constexpr long long OFF_Z2  = OFF_M2 + (long long)BB*NN;
static constexpr long long OFF_MB  = OFF_Z2 + (long long)BB*NN;   // [2][BB]

// LDS load with compile-time immediate offset, forced onto the DS pipe.
template<int OFS>
__device__ __forceinline__ float lds_load_b32(unsigned addr) {
  float r;
  asm volatile("ds_load_b32 %0, %1 offset:%2" : "=v"(r) : "v"(addr), "i"(OFS));
  return r;
}

// ---------------------------------------------------------------------------
// Kernel 1: projections via V_WMMA_F32_16X16X4_F32.
// z = 0: qk1   z = 1: qk2   z = 2..5: v1 (ct = z-2)   z = 6..9: v2
// A (16x4, M=out-ch) lane<16: M=lane, regs K=4kk+0/1; lane>=16: M=lane-16, K=4kk+2/3
// B (4x16, N=pixel)  lane<16: N=lane, regs K=4kk+0/1; lane>=16: N=lane-16, K=4kk+2/3
// C (16x16) reg r: M = r (+8 for hi half), N = lane&15
// ---------------------------------------------------------------------------
__global__ void __launch_bounds__(32) proj_kernel(
    const float* __restrict__ x1, const float* __restrict__ x2,
    const float* __restrict__ Wqk1, const float* __restrict__ bqk1,
    const float* __restrict__ Wqk2, const float* __restrict__ bqk2,
    const float* __restrict__ Wv1,  const float* __restrict__ bv1,
    const float* __restrict__ Wv2,  const float* __restrict__ bv2,
    float* __restrict__ ws)
{
  const int lane = threadIdx.x;
  const bool hi  = lane >= 16;
  const int l15  = lane & 15;
  const int j0   = blockIdx.x * 16;
  const int b    = blockIdx.y;
  const int z    = blockIdx.z;

  const float* X; const float* W; const float* bias;
  int m0 = 0;
  if (z == 0)      { X = x1 + (long long)b*CC*NN; W = Wqk1; bias = bqk1; }
  else if (z == 1) { X = x2 + (long long)b*CC*NN; W = Wqk2; bias = bqk2; }
  else {
    int zz = z - 2;
    int which = zz >> 2;
    m0 = (zz & 3) * 16;
    X    = (which == 0 ? x1 : x2) + (long long)b*CC*NN;
    W    = (which == 0 ? Wv1 : Wv2);
    bias = (which == 0 ? bv1 : bv2);
  }

  const int jme = j0 + l15;
  v8f acc = {0.f,0.f,0.f,0.f,0.f,0.f,0.f,0.f};
#pragma unroll
  for (int kk = 0; kk < 16; ++kk) {
    const int k0 = kk*4 + (hi ? 2 : 0);
    v2f A, Bv;
    A.x  = W[(m0 + l15)*CC + k0];
    A.y  = W[(m0 + l15)*CC + k0 + 1];
    Bv.x = X[(long long)k0*NN + jme];
    Bv.y = X[(long long)(k0+1)*NN + jme];
    acc = __builtin_amdgcn_wmma_f32_16x16x4_f32(false, A, false, Bv,
                                                (short)0, acc, false, false);
  }

  if (z < 2) {
    // rows 0..7 = q channels, rows 8..15 = k channels; store transposed [n][8]
    float* qT = ws + (z==0 ? OFF_QT1 : OFF_QT2) + (long long)b*NN*CRr;
    float* kT = ws + (z==0 ? OFF_KT1 : OFF_KT2) + (long long)b*NN*CRr;
    float* dst = hi ? kT : qT;
#pragma unroll
    for (int r = 0; r < 8; ++r)
      dst[(long long)jme*CRr + r] = acc[r] + bias[r + (hi ? 8 : 0)];
  } else {
    int which = (z-2) >> 2;
    float* vT = ws + (which==0 ? OFF_VT1 : OFF_VT2) + (long long)b*NN*CC;
#pragma unroll
    for (int r = 0; r < 8; ++r) {
      int c = m0 + r + (hi ? 8 : 0);
      vT[(long long)jme*CC + c] = acc[r] + bias[c];
    }
  }
}

// ---------------------------------------------------------------------------
// Kernel 2: per-row softmax stats (m_i, Z_i).  lane = row i; K=8 dot per j.
// ---------------------------------------------------------------------------
__global__ void __launch_bounds__(32) stats_kernel(float* __restrict__ ws)
{
  const int lane = threadIdx.x;
  const int i = blockIdx.x*32 + lane;
  const int b = blockIdx.y;
  const int a = blockIdx.z;  // 0: attn1 (q1,k1)  1: attn2 (q2,k2)
  const float* qT = ws + (a==0 ? OFF_QT1 : OFF_QT2) + (long long)b*NN*CRr;
  const float* kT = ws + (a==0 ? OFF_KT1 : OFF_KT2) + (long long)b*NN*CRr;

  const float4 qa = *(const float4*)(qT + (long long)i*CRr);
  const float4 qb = *(const float4*)(qT + (long long)i*CRr + 4);

  float m = -3.4e38f;
  for (int j = 0; j < NN; ++j) {
    const float4 ka = *(const float4*)(kT + (long long)j*CRr);
    const float4 kb = *(const float4*)(kT + (long long)j*CRr + 4);
    float s = qa.x*ka.x + qa.y*ka.y + qa.z*ka.z + qa.w*ka.w
            + qb.x*kb.x + qb.y*kb.y + qb.z*kb.z + qb.w*kb.w;
    m = fmaxf(m, s);
  }
  float Z = 0.f;
  for (int j = 0; j < NN; ++j) {
    const float4 ka = *(const float4*)(kT + (long long)j*CRr);
    const float4 kb = *(const float4*)(kT + (long long)j*CRr + 4);
    float s = qa.x*ka.x + qa.y*ka.y + qa.z*ka.z + qa.w*ka.w
            + qb.x*kb.x + qb.y*kb.y + qb.z*kb.z + qb.w*kb.w;
    Z += __expf(s - m);
  }
  ws[(a==0?OFF_M1:OFF_M2) + (long long)b*NN + i] = m;
  ws[(a==0?OFF_Z1:OFF_Z2) + (long long)b*NN + i] = Z;
}

// ---------------------------------------------------------------------------
// Kernel 3: per-batch global score max (for stable exp without per-row work).
// ---------------------------------------------------------------------------
__global__ void __launch_bounds__(32) rowmax_kernel(float* __restrict__ ws)
{
  const int a = blockIdx.x, b = blockIdx.y, lane = threadIdx.x;
  const float* m = ws + (a==0?OFF_M1:OFF_M2) + (long long)b*NN;
  float mx = -3.4e38f;
  for (int i = lane; i < NN; i += 32) mx = fmaxf(mx, m[i]);
  for (int off = 16; off > 0; off >>= 1) mx = fmaxf(mx, __shfl_xor(mx, off, 32));
  if (lane == 0) ws[OFF_MB + a*BB + b] = mx;
}

// ---------------------------------------------------------------------------
// Kernel 4: fold g_i = exp(Mb - m_i)/Z_i into columns of v (in place).
// vT1 pairs with attn2 stats (o1 = v1 x attn2); vT2 with attn1.
// ---------------------------------------------------------------------------
__global__ void __launch_bounds__(256) scalev_kernel(float* __restrict__ ws)
{
  const long long e = (long long)blockIdx.x*256 + threadIdx.x;
  const int which = (int)(e / ((long long)BB*NN*CC));      // 0: vT1, 1: vT2
  const long long rem = e - (long long)which*BB*NN*CC;
  const int b = (int)(rem / ((long long)NN*CC));
  const int i = (int)((rem / CC) % NN);
  const int a = which ^ 1;                                  // stats index
  const float m  = ws[(a==0?OFF_M1:OFF_M2) + (long long)b*NN + i];
  const float Z  = ws[(a==0?OFF_Z1:OFF_Z2) + (long long)b*NN + i];
  const float Mb = ws[OFF_MB + a*BB + b];
  const float g  = __expf(Mb - m) / Z;
  ws[(which==0?OFF_VT1:OFF_VT2) + rem] *= g;
}

// ---------------------------------------------------------------------------
// Kernel 5: fused  o = vtilde x exp(S - Mb)  + residual, scaled by gamma/beta.
// Block = 4 waves; each wave owns a 16-col j-tile and a 64x16 fp32 accumulator.
// v-tiles are double-buffered in LDS via GLOBAL_LOAD_ASYNC_TO_LDS_B128
// (ASYNCcnt-tracked); A-operands come back out through explicit ds_load_b32
// (DScnt-tracked), so LDS traffic really uses the DS pipe. Scores are built
// in WMMA-B layout (VALU dot8 + v_exp_f32) feeding v_wmma_f32_16x16x4_f32.
// ---------------------------------------------------------------------------
__global__ void __launch_bounds__(128) attn_out_kernel(
    const float* __restrict__ x1, const float* __restrict__ x2,
    const float* __restrict__ gamma, const float* __restrict__ beta,
    const float* __restrict__ ws, float* __restrict__ out)
{
  __shared__ float vs[2*16*CC];         // 2 x 4KB double-buffered v tiles
  const int tid  = threadIdx.x;
  const int lane = tid & 31;
  const int wave = tid >> 5;
  const bool hi  = lane >= 16;
  const int l15  = lane & 15;
  const int b = blockIdx.y;
  const int z = blockIdx.z;             // 0 -> o1 (attn2, vt1), 1 -> o2
  const int a = (z==0) ? 1 : 0;
  const int j0 = (blockIdx.x*4 + wave)*16;

  const float* qT = ws + (a==0?OFF_QT1:OFF_QT2) + (long long)b*NN*CRr;
  const float* kT = ws + (a==0?OFF_KT1:OFF_KT2) + (long long)b*NN*CRr;
  const float* vT = ws + (z==0?OFF_VT1:OFF_VT2) + (long long)b*NN*CC;
  const float  Mb = ws[OFF_MB + a*BB + b];
  const float  sc = (z==0) ? gamma[0] : beta[0];
  const float* xr = (z==0 ? x1 : x2) + (long long)b*CC*NN;
  float*       op = out + (long long)z*BB*CC*NN + (long long)b*CC*NN;

  // Visible init of vs: keeps the LDS allocation alive/written in IR; the
  // barrier's DScnt wait orders these stores before the async writes below.
  {
    float4 zero4 = make_float4(0.f, 0.f, 0.f, 0.f);
    float4* d4 = (float4*)vs;
#pragma unroll
    for (int q = 0; q < 4; ++q) d4[q*128 + tid] = zero4;   // 512 float4 total
  }
  __syncthreads();

  // k row for my output column, reused for all 2304 score rows
  const int jme = j0 + l15;
  const float4 ka = *(const float4*)(kT + (long long)jme*CRr);
  const float4 kb = *(const float4*)(kT + (long long)jme*CRr + 4);

  v8f acc0 = {0.f,0.f,0.f,0.f,0.f,0.f,0.f,0.f};
  v8f acc1 = acc0, acc2 = acc0, acc3 = acc0;

  // prologue: async-load tile 0 into LDS buffer 0.
  // INST_OFFSET is added to BOTH the global and LDS addresses, so one
  // offset:2048 instruction covers the second half of the 4KB tile.
  {
    const char* g = (const char*)vT + tid*16;
    unsigned    l = (unsigned)(tid*16);          // vs is at LDS offset 0
    asm volatile("global_load_async_to_lds_b128 %0, %1, off"
                 :: "v"(l), "v"(g) : "memory");
    asm volatile("global_load_async_to_lds_b128 %0, %1, off offset:2048"
                 :: "v"(l), "v"(g) : "memory");
  }

  for (int it = 0; it < NN/16; ++it) {
    const int i0 = it*16;
    const int p  = it & 1;
    // my async loads for tile `it` done -> barrier publishes the buffer and
    // guarantees everyone finished computing from the other buffer.
    asm volatile("s_wait_asynccnt 0x0" ::: "memory");
    __syncthreads();

    if (it + 1 < NN/16) {                // issue next tile into other buffer
      const char* g = (const char*)(vT + (long long)(i0+16)*CC) + tid*16;
      unsigned    l = (unsigned)((p ^ 1)*4096 + tid*16);
      asm volatile("global_load_async_to_lds_b128 %0, %1, off"
                   :: "v"(l), "v"(g) : "memory");
      asm volatile("global_load_async_to_lds_b128 %0, %1, off offset:2048"
                   :: "v"(l), "v"(g) : "memory");
    }

    // per-lane LDS base: buffer + my column + (hi ? row+2 : row+0)*256B
    const unsigned ldsb = (unsigned)(p*4096 + l15*4 + (hi ? 512 : 0));

#pragma unroll
    for (int t = 0; t < 4; ++t) {
      // B-operand rows for this K-step: lo half K=4t+0/1, hi half K=4t+2/3
      const int r0 = 4*t + (hi ? 2 : 0);
      const float* q0 = qT + (long long)(i0 + r0)*CRr;
      const float4 qa0 = *(const float4*)q0;
      const float4 qb0 = *(const float4*)(q0 + 4);
      const float4 qa1 = *(const float4*)(q0 + CRr);
      const float4 qb1 = *(const float4*)(q0 + CRr + 4);
      const float s0 = qa0.x*ka.x + qa0.y*ka.y + qa0.z*ka.z + qa0.w*ka.w
                     + qb0.x*kb.x + qb0.y*kb.y + qb0.z*kb.z + qb0.w*kb.w;
      const float s1 = qa1.x*ka.x + qa1.y*ka.y + qa1.z*ka.z + qa1.w*ka.w
                     + qb1.x*kb.x + qb1.y*kb.y + qb1.z*kb.z + qb1.w*kb.w;
      v2f P;
      P.x = __expf(s0 - Mb);
      P.y = __expf(s1 - Mb);

      // A-operands: vtilde[row r0 / r0+1][ct*16 + l15] via ds_load_b32.
      // runtime addr carries t*1024 (row group); immediates carry ct*64 and
      // +256 (row+1).
      const unsigned ad = ldsb + (unsigned)(t*1024);
      float a00 = lds_load_b32<  0>(ad);
      float a01 = lds_load_b32<256>(ad);
      float a10 = lds_load_b32< 64>(ad);
      float a11 = lds_load_b32<320>(ad);
      float a20 = lds_load_b32<128>(ad);
      float a21 = lds_load_b32<384>(ad);
      float a30 = lds_load_b32<192>(ad);
      float a31 = lds_load_b32<448>(ad);
      asm volatile("s_wait_dscnt 0x0" ::: "memory");

      v2f A;
      A.x = a00; A.y = a01;
      acc0 = __builtin_amdgcn_wmma_f32_16x16x4_f32(false, A, false, P, (short)0, acc0, false, false);
      A.x = a10; A.y = a11;
      acc1 = __builtin_amdgcn_wmma_f32_16x16x4_f32(false, A, false, P, (short)0, acc1, false, false);
      A.x = a20; A.y = a21;
      acc2 = __builtin_amdgcn_wmma_f32_16x16x4_f32(false, A, false, P, (short)0, acc2, false, false);
      A.x = a30; A.y = a31;
      acc3 = __builtin_amdgcn_wmma_f32_16x16x4_f32(false, A, false, P, (short)0, acc3, false, false);
    }
  }

  // epilogue: C-layout reg r -> channel ct*16 + r (+8 for hi half), col = jme
#pragma unroll
  for (int ct = 0; ct < 4; ++ct) {
    v8f acc = (ct==0) ? acc0 : (ct==1) ? acc1 : (ct==2) ? acc2 : acc3;
#pragma unroll
    for (int r = 0; r < 8; ++r) {
      const int c = ct*16 + r + (hi ? 8 : 0);
      const long long idx = (long long)c*NN + jme;
      op[idx] = acc[r]*sc + xr[idx];
    }
  }
}

// ---------------------------------------------------------------------------
extern "C" void kernel_launch(void* const* d_in, const int* in_sizes, int n_in,
                              void* d_out, int out_size, void* d_ws, size_t ws_size,
                              hipStream_t stream) {
  const float* x1   = (const float*)d_in[0];
  const float* x2   = (const float*)d_in[1];
  const float* Wqk1 = (const float*)d_in[2];
  const float* bqk1 = (const float*)d_in[3];
  const float* Wqk2 = (const float*)d_in[4];
  const float* bqk2 = (const float*)d_in[5];
  const float* Wv1  = (const float*)d_in[6];
  const float* bv1  = (const float*)d_in[7];
  const float* Wv2  = (const float*)d_in[8];
  const float* bv2  = (const float*)d_in[9];
  const float* gamma = (const float*)d_in[10];
  const float* beta  = (const float*)d_in[11];
  float* ws  = (float*)d_ws;
  float* out = (float*)d_out;

  proj_kernel<<<dim3(NN/16, BB, 10), 32, 0, stream>>>(
      x1, x2, Wqk1, bqk1, Wqk2, bqk2, Wv1, bv1, Wv2, bv2, ws);
  stats_kernel<<<dim3(NN/32, BB, 2), 32, 0, stream>>>(ws);
  rowmax_kernel<<<dim3(2, BB), 32, 0, stream>>>(ws);
  scalev_kernel<<<dim3((unsigned)((2LL*BB*NN*CC)/256)), 256, 0, stream>>>(ws);
  attn_out_kernel<<<dim3(NN/64, BB, 2), 128, 0, stream>>>(
      x1, x2, gamma, beta, ws, out);
}